// MaskedResizer_85461259255929
// MI455X (gfx1250) — compile-verified
//
#include <hip/hip_runtime.h>
#include <hip/hip_bf16.h>
#include <stdint.h>

#define B_  32
#define C_  256
#define T_  8192
#define TO_ 4096   // T/2

#define TJ   256   // j-tile per block
#define CPB  4     // channels per block
#define SROW 520   // fixed staged window (floats) per channel; need <= 516

// ---------------------------------------------------------------------------
// Phase B (hot kernel, placed first so the disasm snippet shows it):
// fixed 520-float window per channel staged into LDS via CDNA5 async
// global->LDS b128 copies, then irregular lerp gathers from LDS.
// Window bound: scale = valid_len/ceil(valid_len/2) <= 2 exactly
//   => i1(jBase+255) <= i0(jBase) + 512.
// ---------------------------------------------------------------------------
__global__ __launch_bounds__(256)
void mr_phaseB(const float* __restrict__ x,
               const int2* __restrict__ meta,
               float* __restrict__ out) {
    const int b     = blockIdx.z;
    const int cBase = blockIdx.y * CPB;
    const int jBase = blockIdx.x * TJ;
    const int t     = threadIdx.x;
    const int j     = jBase + t;

    __shared__ __align__(16) float stage[CPB][SROW];
    __shared__ int s_lo;

    const int2 m  = meta[(size_t)b * TO_ + j];
    const int  pk = m.x;                    // i0 | i1<<16, or -1
    const float w = __int_as_float(m.y);

    if (t == 0) s_lo = pk;                  // validity is a j-prefix: thread0 = min i0
    __syncthreads();
    const int lov = s_lo;

    if (lov < 0) {                          // whole tile past out_len (uniform branch)
        for (int c = 0; c < CPB; ++c)
            __builtin_nontemporal_store(
                0.0f, &out[((size_t)(b * C_ + cBase + c)) * TO_ + j]);
        return;
    }

    const int lo  = lov & 0xFFFF;
    const int loS = lo & ~3;                // 16B-aligned window start

    // Async contiguous copy: global -> LDS, 16 bytes per active lane per issue.
    // Fixed shape: 130 b128 chunks per channel; global address clamped to the
    // row end (clamped duplicates land in LDS slots that are never gathered).
    constexpr int nVec = SROW / 4;          // 130
    const int maxOff = T_ - 4;
#pragma unroll 1
    for (int it = t; it < nVec * CPB; it += TJ) {
        const int c   = it / nVec;          // division by constant
        const int v   = it - c * nVec;
        const int off = min(loS + v * 4, maxOff);
        const float* gptr = x + ((size_t)(b * C_ + cBase + c)) * T_ + off;
        uint32_t ldsAddr = (uint32_t)(uintptr_t)&stage[c][v * 4];
        asm volatile("global_load_async_to_lds_b128 %0, %1, off"
                     :: "v"(ldsAddr), "v"((uint64_t)(uintptr_t)gptr)
                     : "memory");
    }
    __builtin_amdgcn_s_wait_asynccnt(0);  // this wave's async copies landed in LDS
    __syncthreads();                      // all waves' copies visible

    const bool  valid = (pk >= 0);
    const float w1    = 1.0f - w;
    const int   o0    = (pk & 0xFFFF) - loS;
    const int   o1    = (pk >> 16) - loS;     // pk >= 0 when used
    for (int c = 0; c < CPB; ++c) {
        float r = 0.0f;
        if (valid) {
            float g0 = stage[c][o0];
            float g1 = stage[c][o1];
            r = w1 * g0 + w * g1;
        }
        __builtin_nontemporal_store(
            r, &out[((size_t)(b * C_ + cBase + c)) * TO_ + j]);
    }
}

// ---------------------------------------------------------------------------
// Phase A: per-batch mask reductions + per-(b,j) resample metadata.
// mask bytes are 0/1 -> popc(dword) counts padded bytes.
// meta[b*TO_+j] = { i0 | (i1<<16)  (or -1 if j >= out_len), bitcast(w) }
// Also writes mask_out (as float 0/1) into the tail of d_out.
// ---------------------------------------------------------------------------
__global__ __launch_bounds__(256)
void mr_phaseA(const uint4* __restrict__ mask4,
               float* __restrict__ mask_out_f,
               int2* __restrict__ meta) {
    const int b = blockIdx.x;
    const int t = threadIdx.x;
    const uint4* mrow = mask4 + (size_t)b * (T_ / 16);
    float* morow = mask_out_f + (size_t)b * TO_;

    __shared__ int red[256];

    int cnt = 0;   // valid_len contribution
    int cnt2 = 0;  // out_len contribution
#pragma unroll
    for (int k = 0; k < T_ / 16 / 256; ++k) {   // 2 iterations
        const int idx = t + k * 256;            // 16-byte chunk -> 8 output pairs
        uint4 v = mrow[idx];
        cnt += 16 - (__popc(v.x) + __popc(v.y) + __popc(v.z) + __popc(v.w));
        uint32_t arr[4] = {v.x, v.y, v.z, v.w};
        float* mo = morow + idx * 8;
#pragma unroll
        for (int q = 0; q < 4; ++q) {
            uint32_t u = arr[q];
            int m0 = (u & (u >> 8)) & 1;           // mask[2j] && mask[2j+1]
            int m1 = ((u >> 16) & (u >> 24)) & 1;
            mo[q * 2 + 0] = (float)m0;
            mo[q * 2 + 1] = (float)m1;
            cnt2 += 2 - m0 - m1;
        }
    }

    // joint reduction: cnt <= 8192, cnt2 <= 4096 -> pack into one int
    red[t] = cnt | (cnt2 << 16);
    __syncthreads();
    for (int s = 128; s > 0; s >>= 1) {
        if (t < s) red[t] += red[t + s];
        __syncthreads();
    }
    const int packed    = red[0];
    const int valid_len = packed & 0xFFFF;
    const int out_len   = packed >> 16;

    const int   last  = max(valid_len - 1, 0);
    const float scale = (float)valid_len / (float)max(out_len, 1);

    for (int j = t; j < TO_; j += 256) {
        float src = fmaxf(((float)j + 0.5f) * scale - 0.5f, 0.0f);
        float fl  = floorf(src);
        float w   = src - fl;
        int   i0  = min((int)fl, last);     // src >= 0 so floor >= 0 already
        int   i1  = min(i0 + 1, last);
        int2 mm;
        mm.x = (j < out_len) ? (i0 | (i1 << 16)) : -1;
        mm.y = __float_as_int(w);
        meta[(size_t)b * TO_ + j] = mm;
    }
}

// ---------------------------------------------------------------------------
extern "C" void kernel_launch(void* const* d_in, const int* in_sizes, int n_in,
                              void* d_out, int out_size, void* d_ws, size_t ws_size,
                              hipStream_t stream) {
    (void)in_sizes; (void)n_in; (void)out_size; (void)ws_size;

    const float* x     = (const float*)d_in[0];
    const uint4* mask4 = (const uint4*)d_in[1];   // bool mask, 1 byte/elem, 16B chunks

    float* out        = (float*)d_out;                 // [B, C, TO]
    float* mask_out_f = out + (size_t)B_ * C_ * TO_;   // [B, TO] as float 0/1

    int2* meta = (int2*)d_ws;                          // [B*TO] {i0|i1<<16, w}

    mr_phaseA<<<B_, 256, 0, stream>>>(mask4, mask_out_f, meta);

    dim3 grid(TO_ / TJ, C_ / CPB, B_);
    mr_phaseB<<<grid, TJ, 0, stream>>>(x, meta, out);
}